// PathQualityAwareConv_30734785970663
// MI455X (gfx1250) — compile-verified
//
#include <hip/hip_runtime.h>
#include <hip/hip_bf16.h>

#define NN 50000
#define EE 800000

typedef __attribute__((ext_vector_type(2))) float v2f;
typedef __attribute__((ext_vector_type(8))) float v8f;

__device__ __forceinline__ unsigned ford(float f) {
    unsigned u = __float_as_uint(f);
    return (u & 0x80000000u) ? ~u : (u | 0x80000000u);
}
__device__ __forceinline__ float ford_inv(unsigned o) {
    return __uint_as_float((o & 0x80000000u) ? (o & 0x7FFFFFFFu) : ~o);
}

// ---------------- zero init of accumulators ----------------
__global__ void k_zero(float* conv, unsigned* mmax, float* den) {
    int i = blockIdx.x * blockDim.x + threadIdx.x;
    if (i < NN * 64) conv[i] = 0.0f;
    if (i < NN * 4) { mmax[i] = 0u; den[i] = 0.0f; }
}

// ---------------- fold small matrices on device ----------------
// A_edge[k][h] = sum_c W_edge[k, h*64+c] * att_edge[h,c]
// B2[i][h]    = sum_k Wq2[i,k] * A_edge[k][h]
// cb[h]       = sum_k bq2[k]  * A_edge[k][h]
// cvec[c]     = bc[c] + sum_{j<8} net[j]*Wc[64+j,c] + sum_k bias[k]*Wc[k,c]
__global__ void k_precompute(const float* __restrict__ W_edge, const float* __restrict__ att_edge,
                             const float* __restrict__ Wq2, const float* __restrict__ bq2,
                             const float* __restrict__ Wc, const float* __restrict__ bc,
                             const float* __restrict__ net, const float* __restrict__ bias,
                             float* A_edge, float* B2, float* cb, float* cvec) {
    __shared__ float sA[64 * 4];
    int t = threadIdx.x;          // 256 threads
    {
        int k = t >> 2, h = t & 3;
        float s = 0.f;
        for (int c = 0; c < 64; ++c) s += W_edge[k * 256 + h * 64 + c] * att_edge[h * 64 + c];
        sA[k * 4 + h] = s;
        A_edge[k * 4 + h] = s;
    }
    __syncthreads();
    if (t < 128) {
        int i = t >> 2, h = t & 3;
        float s = 0.f;
        for (int k = 0; k < 64; ++k) s += Wq2[i * 64 + k] * sA[k * 4 + h];
        B2[i * 4 + h] = s;
    }
    if (t < 4) {
        float s = 0.f;
        for (int k = 0; k < 64; ++k) s += bq2[k] * sA[k * 4 + t];
        cb[t] = s;
    }
    if (t < 64) {
        float s = bc[t];
        for (int j = 0; j < 8; ++j) s += net[j] * Wc[(64 + j) * 64 + t];
        for (int k = 0; k < 64; ++k) s += bias[k] * Wc[k * 64 + t];
        cvec[t] = s;
    }
}

// ---------------- h = x @ W via V_WMMA_F32_16X16X4_F32 ----------------
// one wave -> one 16x16 tile of h[N,256]; K=128 in 32 steps of 4
__global__ __launch_bounds__(256) void k_gemm_h(const float* __restrict__ x,
                                                const float* __restrict__ W,
                                                float* __restrict__ h) {
    int wave = (blockIdx.x * 256 + threadIdx.x) >> 5;
    int lane = threadIdx.x & 31;
    int row0 = (wave >> 4) * 16;   // 3125 row tiles
    int col0 = (wave & 15) * 16;   // 16 col tiles
    int lm = lane & 15;
    int kk = (lane >> 4) * 2;      // 0 or 2: K sub-offset for this lane half
    v8f c = {};
    const float* xrow = x + (size_t)(row0 + lm) * 128 + kk;
    const float* wcol = W + (size_t)kk * 256 + col0 + lm;
#pragma unroll 4
    for (int kc = 0; kc < 32; ++kc) {
        v2f a = *(const v2f*)(xrow + kc * 4);          // A: K=kk, kk+1
        v2f b;
        b[0] = wcol[(size_t)kc * 4 * 256];             // B: K=kk
        b[1] = wcol[(size_t)kc * 4 * 256 + 256];       // B: K=kk+1
        c = __builtin_amdgcn_wmma_f32_16x16x4_f32(false, a, false, b, (short)0, c, false, false);
    }
    int mbase = row0 + (lane >> 4) * 8;
    float* out = h + col0 + lm;
#pragma unroll
    for (int v = 0; v < 8; ++v) out[(size_t)(mbase + v) * 256] = c[v];
}

// ---------------- per-node attention logits ----------------
__global__ void k_node_att(const float* __restrict__ h, const float* __restrict__ att_src,
                           const float* __restrict__ att_dst, float* a_src, float* a_dst) {
    int idx = blockIdx.x * blockDim.x + threadIdx.x;
    if (idx >= NN * 4) return;
    int n = idx >> 2, hh = idx & 3;
    const float* hp = h + (size_t)n * 256 + hh * 64;
    float as = 0.f, ad = 0.f;
    for (int cc = 0; cc < 64; ++cc) {
        float v = hp[cc];
        as += v * att_src[hh * 64 + cc];
        ad += v * att_dst[hh * 64 + cc];
    }
    a_src[idx] = as;
    a_dst[idx] = ad;
}

// ---------------- per-edge alpha + segment max ----------------
__global__ __launch_bounds__(256) void k_edge_alpha(const int* __restrict__ ei,
                                                    const float* __restrict__ edge_attr,
                                                    const float* __restrict__ pq,
                                                    const float* __restrict__ Wq1,
                                                    const float* __restrict__ bq1,
                                                    const float* __restrict__ A_edge,
                                                    const float* __restrict__ B2,
                                                    const float* __restrict__ cb,
                                                    const float* __restrict__ a_src,
                                                    const float* __restrict__ a_dst,
                                                    float* alpha, unsigned* mmax) {
    __shared__ float sA[256], sB[128], sc[4];
    int t = threadIdx.x;
    sA[t] = A_edge[t];
    if (t < 128) sB[t] = B2[t];
    if (t < 4) sc[t] = cb[t];
    __syncthreads();
    int e = blockIdx.x * 256 + t;
    if (e >= EE) return;
    int s = ei[e], d = ei[EE + e];
    float q0 = pq[e * 4 + 0] * 0.01f, q1 = pq[e * 4 + 1] * 0.01f;
    float q2 = pq[e * 4 + 2] * 100.f, q3 = pq[e * 4 + 3] * 100.f;
    float ae[4];
#pragma unroll
    for (int hh = 0; hh < 4; ++hh) ae[hh] = sc[hh];
    for (int i = 0; i < 32; ++i) {
        float hid = fmaf(q0, Wq1[i], fmaf(q1, Wq1[32 + i], fmaf(q2, Wq1[64 + i], fmaf(q3, Wq1[96 + i], bq1[i]))));
        hid = fmaxf(hid, 0.f);
#pragma unroll
        for (int hh = 0; hh < 4; ++hh) ae[hh] += hid * sB[i * 4 + hh];
    }
    const float* ea = edge_attr + (size_t)e * 64;
    for (int k = 0; k < 64; ++k) {
        float v = ea[k];
#pragma unroll
        for (int hh = 0; hh < 4; ++hh) ae[hh] += v * sA[k * 4 + hh];
    }
#pragma unroll
    for (int hh = 0; hh < 4; ++hh) {
        float al = a_src[s * 4 + hh] + a_dst[d * 4 + hh] + ae[hh];
        al = (al > 0.f) ? al : 0.2f * al;          // leaky relu
        alpha[e * 4 + hh] = al;
        atomicMax(&mmax[d * 4 + hh], ford(al));
    }
}

// ---------------- exp + segment sum ----------------
__global__ void k_edge_exp(const int* __restrict__ ei, const unsigned* __restrict__ mmax,
                           float* alpha_ex, float* den) {
    int idx = blockIdx.x * blockDim.x + threadIdx.x;
    if (idx >= EE * 4) return;
    int e = idx >> 2, hh = idx & 3;
    int d = ei[EE + e];
    float m = ford_inv(mmax[d * 4 + hh]);
    float ex = __expf(alpha_ex[idx] - m);
    alpha_ex[idx] = ex;
    atomicAdd(&den[d * 4 + hh], ex);
}

// ---------------- weighted scatter (head-mean folded in) ----------------
__global__ void k_edge_scatter(const int* __restrict__ ei, const float* __restrict__ h,
                               const float* __restrict__ exw, const float* __restrict__ den,
                               float* conv) {
    int idx = blockIdx.x * blockDim.x + threadIdx.x;   // EE*64 threads
    int e = idx >> 6, cc = idx & 63;
    if (e >= EE) return;
    int s = ei[e], d = ei[EE + e];
    float acc = 0.f;
#pragma unroll
    for (int hh = 0; hh < 4; ++hh) {
        float w = exw[e * 4 + hh] / den[d * 4 + hh];
        acc += w * h[(size_t)s * 256 + hh * 64 + cc];
    }
    atomicAdd(&conv[d * 64 + cc], 0.25f * acc);
}

// ---------------- out = relu(conv @ Wc_top + cvec) via WMMA ----------------
__global__ __launch_bounds__(256) void k_gemm_out(const float* __restrict__ conv,
                                                  const float* __restrict__ Wc,
                                                  const float* __restrict__ cvec,
                                                  float* __restrict__ out) {
    int wave = (blockIdx.x * 256 + threadIdx.x) >> 5;
    if (wave >= 3125 * 4) return;
    int lane = threadIdx.x & 31;
    int row0 = (wave >> 2) * 16;
    int col0 = (wave & 3) * 16;
    int lm = lane & 15;
    int kk = (lane >> 4) * 2;
    float cv = cvec[col0 + lm];
    v8f c;
#pragma unroll
    for (int v = 0; v < 8; ++v) c[v] = cv;
    const float* arow = conv + (size_t)(row0 + lm) * 64 + kk;
    const float* brow = Wc + (size_t)kk * 64 + col0 + lm;
#pragma unroll 4
    for (int kc = 0; kc < 16; ++kc) {
        v2f a = *(const v2f*)(arow + kc * 4);
        v2f b;
        b[0] = brow[(size_t)kc * 4 * 64];
        b[1] = brow[(size_t)kc * 4 * 64 + 64];
        c = __builtin_amdgcn_wmma_f32_16x16x4_f32(false, a, false, b, (short)0, c, false, false);
    }
    int mbase = row0 + (lane >> 4) * 8;
    float* op = out + col0 + lm;
#pragma unroll
    for (int v = 0; v < 8; ++v) op[(size_t)(mbase + v) * 64] = fmaxf(c[v], 0.f);
}

extern "C" void kernel_launch(void* const* d_in, const int* in_sizes, int n_in,
                              void* d_out, int out_size, void* d_ws, size_t ws_size,
                              hipStream_t stream) {
    const float* x         = (const float*)d_in[0];
    const int*   ei        = (const int*)d_in[1];
    const float* edge_attr = (const float*)d_in[2];
    const float* pq        = (const float*)d_in[3];
    const float* net       = (const float*)d_in[4];
    const float* W         = (const float*)d_in[5];
    const float* att_src   = (const float*)d_in[6];
    const float* att_dst   = (const float*)d_in[7];
    const float* W_edge    = (const float*)d_in[8];
    const float* att_edge  = (const float*)d_in[9];
    const float* bias      = (const float*)d_in[10];
    const float* Wq1       = (const float*)d_in[11];
    const float* bq1       = (const float*)d_in[12];
    const float* Wq2       = (const float*)d_in[13];
    const float* bq2       = (const float*)d_in[14];
    const float* Wc        = (const float*)d_in[15];
    const float* bc        = (const float*)d_in[16];
    float* out = (float*)d_out;

    // workspace layout (floats)
    float*    h      = (float*)d_ws;                       // NN*256
    float*    conv   = h + (size_t)NN * 256;               // NN*64
    float*    a_src  = conv + (size_t)NN * 64;             // NN*4
    float*    a_dst  = a_src + (size_t)NN * 4;             // NN*4
    unsigned* mmax   = (unsigned*)(a_dst + (size_t)NN * 4);// NN*4
    float*    den    = (float*)(mmax + (size_t)NN * 4);    // NN*4
    float*    exw    = den + (size_t)NN * 4;               // EE*4
    float*    A_edge = exw + (size_t)EE * 4;               // 256
    float*    B2     = A_edge + 256;                       // 128
    float*    cb     = B2 + 128;                           // 4
    float*    cvec   = cb + 4;                             // 64

    k_zero<<<(NN * 64 + 255) / 256, 256, 0, stream>>>(conv, mmax, den);
    k_precompute<<<1, 256, 0, stream>>>(W_edge, att_edge, Wq2, bq2, Wc, bc, net, bias,
                                        A_edge, B2, cb, cvec);
    // h = x @ W : 3125*16 tiles, 8 waves / block
    k_gemm_h<<<(3125 * 16) / 8, 256, 0, stream>>>(x, W, h);
    k_node_att<<<(NN * 4 + 255) / 256, 256, 0, stream>>>(h, att_src, att_dst, a_src, a_dst);
    k_edge_alpha<<<(EE + 255) / 256, 256, 0, stream>>>(ei, edge_attr, pq, Wq1, bq1,
                                                       A_edge, B2, cb, a_src, a_dst, exw, mmax);
    k_edge_exp<<<(EE * 4 + 255) / 256, 256, 0, stream>>>(ei, mmax, exw, den);
    k_edge_scatter<<<(EE * 64 + 255) / 256, 256, 0, stream>>>(ei, h, exw, den, conv);
    // out = relu(conv @ Wc_top + cvec) : 3125*4 tiles
    k_gemm_out<<<(3125 * 4 * 32 + 255) / 256, 256, 0, stream>>>(conv, Wc, cvec, out);
}